// SparseOptimizer_30477087932493
// MI455X (gfx1250) — compile-verified
//
#include <hip/hip_runtime.h>
#include <stdint.h>

// Sparse Adam (elementwise, memory-bound). Single fused streaming pass:
//   read 384MB + 32MB grad + 4MB inv, write 384MB  -> ~35us @ 23.3 TB/s.
// CDNA5 paths used: global_load_async_to_lds_b128 double-buffering (ASYNCcnt),
// s_wait_asynccnt / s_wait_dscnt split counters, b128 NT stores, wave32.
// WMMA is deliberately not used: the op has no contraction; it is pure HBM
// streaming and the matrix pipes cannot contribute.

typedef float v4f __attribute__((ext_vector_type(4)));

#define ADAM_LR   1.6e-4f
#define ADAM_EPS  1e-15f

// ---- CDNA5 async copy helpers (inline asm) ----------------------------------

__device__ __forceinline__ void async_ld16(const v4f* gsrc, const v4f* lds_dst) {
  // VDST = LDS byte offset (low 32 bits of the flat shared pointer are the
  // allocation-relative LDS offset on gfx1250), VADDR = 64-bit global address.
  asm volatile("global_load_async_to_lds_b128 %0, %1, off"
               :: "v"((uint32_t)(uintptr_t)lds_dst), "v"(gsrc)
               : "memory");
}
__device__ __forceinline__ void wait_async_le3() {
  asm volatile("s_wait_asynccnt 0x3" ::: "memory");
}
__device__ __forceinline__ void wait_async_0() {
  asm volatile("s_wait_asynccnt 0x0" ::: "memory");
}
__device__ __forceinline__ void wait_ds_0() {
  asm volatile("s_wait_dscnt 0x0" ::: "memory");
}

// ---- Adam math ---------------------------------------------------------------

__device__ __forceinline__ void bias_terms(int step, float& k, float& rb2) {
  float sf  = (float)step;
  float bc1 = 1.0f - exp2f(sf * log2f(0.9f));    // 0.9^1000 underflows -> bc1=1
  float bc2 = 1.0f - exp2f(sf * log2f(0.999f));
  k   = ADAM_LR / bc1;
  rb2 = 1.0f / sqrtf(bc2);
}

__device__ __forceinline__ void adam_update4(v4f& p, v4f& a, v4f& s, v4f g,
                                             float k, float rb2) {
#pragma unroll
  for (int c = 0; c < 4; ++c) {
    float gg    = g[c];
    float an    = a[c] * 0.9f   + 0.1f   * gg;
    float sn    = s[c] * 0.999f + 0.001f * gg * gg;
    float denom = sqrtf(sn) * rb2 + ADAM_EPS;
    p[c] -= k * (an / denom);
    a[c] = an;
    s[c] = sn;
  }
}

// ---- inverse-index build -----------------------------------------------------

__global__ void inv_init_kernel(int* __restrict__ inv, long n) {
  long i = (long)blockIdx.x * blockDim.x + threadIdx.x;
  if (i < n) inv[i] = -1;
}

__global__ void inv_scatter_kernel(int* __restrict__ inv,
                                   const int* __restrict__ index, long m) {
  long i = (long)blockIdx.x * blockDim.x + threadIdx.x;
  if (i < m) inv[index[i]] = (int)i;
}

// ---- fused streaming kernel --------------------------------------------------
// Tile = 256 float4 = 1024 floats = 32 rows (D=32). Persistent blocks,
// double-buffered async loads for the three streamed arrays.

__global__ __launch_bounds__(256) void sparse_adam_stream(
    const float* __restrict__ param, const float* __restrict__ ea,
    const float* __restrict__ eas,   const float* __restrict__ grad,
    const int* __restrict__ inv,     const int* __restrict__ step_p,
    float* __restrict__ out, long nd, long T) {
  __shared__ v4f smem[2][3][256];   // 24 KB
  const int tid = threadIdx.x;
  const v4f* p4 = (const v4f*)param;
  const v4f* a4 = (const v4f*)ea;
  const v4f* s4 = (const v4f*)eas;

  // Loop-invariant bias corrections: compute once per thread (uniform step).
  float k, rb2;
  bias_terms(*step_p, k, rb2);

  const long stride = (long)gridDim.x;
  long t   = (long)blockIdx.x;
  int  buf = 0;

  if (t < T) {
    long g4 = t * 256 + tid;
    async_ld16(p4 + g4, &smem[0][0][tid]);
    async_ld16(a4 + g4, &smem[0][1][tid]);
    async_ld16(s4 + g4, &smem[0][2][tid]);
  }

  while (t < T) {
    long tn = t + stride;
    if (tn < T) {
      wait_ds_0();                       // prior reads of buf^1 fully retired
      long g4n = tn * 256 + tid;
      async_ld16(p4 + g4n, &smem[buf ^ 1][0][tid]);
      async_ld16(a4 + g4n, &smem[buf ^ 1][1][tid]);
      async_ld16(s4 + g4n, &smem[buf ^ 1][2][tid]);
      wait_async_le3();                  // 3 newest outstanding -> tile t done
    } else {
      wait_async_0();
    }

    long g4 = t * 256 + tid;
    v4f p = smem[buf][0][tid];
    v4f a = smem[buf][1][tid];
    v4f s = smem[buf][2][tid];

    long row = g4 >> 3;                  // 8 float4 per 32-float row
    int  j   = inv ? inv[row] : -1;
    if (j >= 0) {
      v4f g = *(const v4f*)(grad + (size_t)j * 32 + (size_t)(g4 & 7) * 4);
      adam_update4(p, a, s, g, k, rb2);
    }
    // write-once 384MB output: NT hint, keep L2 for the gathered inputs
    __builtin_nontemporal_store(p, (v4f*)out + g4);
    __builtin_nontemporal_store(a, (v4f*)(out + nd) + g4);
    __builtin_nontemporal_store(s, (v4f*)(out + 2 * nd) + g4);

    t = tn;
    buf ^= 1;
  }
}

// Tail for element counts not divisible by 1024 (not hit for N=1e6, D=32).
__global__ void sparse_adam_tail(
    const float* __restrict__ param, const float* __restrict__ ea,
    const float* __restrict__ eas,   const float* __restrict__ grad,
    const int* __restrict__ inv,     const int* __restrict__ step_p,
    float* __restrict__ out, long nd, long base4, long nd4) {
  long g4 = base4 + threadIdx.x;
  if (g4 >= nd4) return;
  v4f p = ((const v4f*)param)[g4];
  v4f a = ((const v4f*)ea)[g4];
  v4f s = ((const v4f*)eas)[g4];
  long row = g4 >> 3;
  int  j   = inv ? inv[row] : -1;
  if (j >= 0) {
    v4f g = *(const v4f*)(grad + (size_t)j * 32 + (size_t)(g4 & 7) * 4);
    float k, rb2;
    bias_terms(*step_p, k, rb2);
    adam_update4(p, a, s, g, k, rb2);
  }
  __builtin_nontemporal_store(p, (v4f*)out + g4);
  __builtin_nontemporal_store(a, (v4f*)(out + nd) + g4);
  __builtin_nontemporal_store(s, (v4f*)(out + 2 * nd) + g4);
}

// Fallback when workspace can't hold inv[N]: overwrite the M updated rows
// after the pure-copy pass (sparse_adam_stream with inv==nullptr).
__global__ void sparse_adam_rows(
    const float* __restrict__ param, const float* __restrict__ ea,
    const float* __restrict__ eas,   const float* __restrict__ grad,
    const int* __restrict__ index,   const int* __restrict__ step_p,
    float* __restrict__ out, long m, long nd) {
  long u = (long)blockIdx.x * blockDim.x + threadIdx.x;
  if (u >= m * 8) return;
  long j    = u >> 3;
  long lane = u & 7;
  long n    = (long)index[j];
  long g4   = n * 8 + lane;
  v4f p = ((const v4f*)param)[g4];
  v4f a = ((const v4f*)ea)[g4];
  v4f s = ((const v4f*)eas)[g4];
  v4f g = ((const v4f*)grad)[j * 8 + lane];
  float k, rb2;
  bias_terms(*step_p, k, rb2);
  adam_update4(p, a, s, g, k, rb2);
  ((v4f*)out)[g4]            = p;
  ((v4f*)(out + nd))[g4]     = a;
  ((v4f*)(out + 2 * nd))[g4] = s;
}

// ---- launch ------------------------------------------------------------------

extern "C" void kernel_launch(void* const* d_in, const int* in_sizes, int n_in,
                              void* d_out, int out_size, void* d_ws, size_t ws_size,
                              hipStream_t stream) {
  const float* param = (const float*)d_in[0];
  const float* grad  = (const float*)d_in[1];
  const float* ea    = (const float*)d_in[2];
  const float* eas   = (const float*)d_in[3];
  const int*   index = (const int*)d_in[4];
  const int*   stepp = (const int*)d_in[5];
  float* out = (float*)d_out;

  const long nd  = (long)in_sizes[0];      // N * D  (D = 32)
  const long md  = (long)in_sizes[1];      // M * D
  const long N   = nd / 32;
  const long M   = md / 32;
  const long nd4 = nd / 4;                 // total float4 groups
  const long T   = nd4 / 256;              // full 256-float4 tiles
  const long R   = nd4 - T * 256;

  const bool use_inv = (d_ws != nullptr) && (ws_size >= (size_t)N * sizeof(int));
  int* inv = use_inv ? (int*)d_ws : nullptr;

  if (use_inv) {
    inv_init_kernel<<<(unsigned)((N + 255) / 256), 256, 0, stream>>>(inv, N);
    if (M > 0)
      inv_scatter_kernel<<<(unsigned)((M + 255) / 256), 256, 0, stream>>>(inv, index, M);
  }

  if (T > 0) {
    long blocks = T < 8192 ? T : 8192;     // persistent, grid-strided tiles
    sparse_adam_stream<<<(unsigned)blocks, 256, 0, stream>>>(
        param, ea, eas, grad, inv, stepp, out, nd, T);
  }
  if (R > 0) {
    sparse_adam_tail<<<1, 256, 0, stream>>>(
        param, ea, eas, grad, inv, stepp, out, nd, T * 256, nd4);
  }
  if (!use_inv && M > 0) {
    long u = M * 8;
    sparse_adam_rows<<<(unsigned)((u + 255) / 256), 256, 0, stream>>>(
        param, ea, eas, grad, index, stepp, out, M, nd);
  }
}